// E8P12_codebook_63874753626531
// MI455X (gfx1250) — compile-verified
//
#include <hip/hip_runtime.h>

typedef __attribute__((ext_vector_type(2))) float v2f;
typedef __attribute__((ext_vector_type(8))) float v8f;

#define NQ        8192      // 256*32 query vectors, dim 8
#define NCODES    65536     // codebook entries, dim 8
#define M_PER_BLOCK 2       // 16-query tiles per block (B-tile reuse)
#define N_TILES   (NCODES / 16)   // 4096
#define WAVES_PER_BLOCK 8
#define NBLOCKS   (NQ / (16 * M_PER_BLOCK))   // 256

// ---------------------------------------------------------------------------
// Kernel 1: fused score-GEMM (WMMA f32 16x16x4, K=8 via 2 chained WMMAs) +
// argmax reduction. score = dot(x,g) - 0.5*||g||^2 (same argmax as reference's
// 2*dot - norm). Each block processes 2 M-tiles against every B-tile it loads,
// halving grid L2 traffic and doubling WMMA per loaded byte.
// ---------------------------------------------------------------------------
__global__ __launch_bounds__(256) void e8p_argmax_wmma(
    const float* __restrict__ X,      // [8192, 8]
    const float* __restrict__ grid,   // [65536, 8]
    const float* __restrict__ gnorm,  // [65536]
    float* __restrict__ vals,         // [8192, 8]
    long long* __restrict__ idxs)     // [8192]
{
    __shared__ float    s_val[M_PER_BLOCK * WAVES_PER_BLOCK * 16];
    __shared__ unsigned s_idx[M_PER_BLOCK * WAVES_PER_BLOCK * 16];

    const int lane = threadIdx.x & 31;
    const int wave = threadIdx.x >> 5;
    const int col  = lane & 15;          // N (for B/C/D) and M (for A)
    const int k0   = (lane >> 4) * 2;    // K pair owned by this half-wave

    const int mt0 = blockIdx.x * M_PER_BLOCK;   // first 16-query tile

    // A operands per M-tile: first WMMA covers K=0..3, second K=4..7
    v2f a01[M_PER_BLOCK], a45[M_PER_BLOCK];
#pragma unroll
    for (int mm = 0; mm < M_PER_BLOCK; ++mm) {
        const int q = (mt0 + mm) * 16 + col;    // A-matrix row M = col
        a01[mm] = *(const v2f*)(X + q * 8 + k0);
        a45[mm] = *(const v2f*)(X + q * 8 + k0 + 4);
    }

    float    best[M_PER_BLOCK][8];
    unsigned bidx[M_PER_BLOCK][8];
#pragma unroll
    for (int mm = 0; mm < M_PER_BLOCK; ++mm)
#pragma unroll
        for (int i = 0; i < 8; ++i) { best[mm][i] = -3.0e38f; bidx[mm][i] = 0u; }

    for (int t = wave; t < N_TILES; t += WAVES_PER_BLOCK) {
        const int n = t * 16 + col;      // codeword column owned by this lane
        const v2f b01 = *(const v2f*)(grid + n * 8 + k0);
        const v2f b45 = *(const v2f*)(grid + n * 8 + k0 + 4);
        const float ci = -0.5f * gnorm[n];
#pragma unroll
        for (int mm = 0; mm < M_PER_BLOCK; ++mm) {
            v8f c = {ci, ci, ci, ci, ci, ci, ci, ci};
            c = __builtin_amdgcn_wmma_f32_16x16x4_f32(false, a01[mm], false, b01,
                                                      (short)0, c, false, false);
            c = __builtin_amdgcn_wmma_f32_16x16x4_f32(false, a45[mm], false, b45,
                                                      (short)0, c, false, false);
#pragma unroll
            for (int i = 0; i < 8; ++i) {
                const bool gt = c[i] > best[mm][i];   // strict >: keeps lowest n
                best[mm][i] = gt ? c[i] : best[mm][i];
                bidx[mm][i] = gt ? (unsigned)n : bidx[mm][i];
            }
        }
    }

    // Butterfly argmax across each 16-lane half (columns of the score tile).
    // Rows 0..7 live in lanes 0..15 (VGPR i = row i), rows 8..15 in lanes 16..31.
#pragma unroll
    for (int off = 8; off >= 1; off >>= 1) {
#pragma unroll
        for (int mm = 0; mm < M_PER_BLOCK; ++mm)
#pragma unroll
            for (int i = 0; i < 8; ++i) {
                const float    ov = __shfl_xor(best[mm][i], off, 32);
                const unsigned oi = (unsigned)__shfl_xor((int)bidx[mm][i], off, 32);
                const bool take = (ov > best[mm][i]) ||
                                  (ov == best[mm][i] && oi < bidx[mm][i]);
                best[mm][i] = take ? ov : best[mm][i];
                bidx[mm][i] = take ? oi : bidx[mm][i];
            }
    }

    if (lane == 0 || lane == 16) {
        const int rbase = (lane == 0) ? 0 : 8;
#pragma unroll
        for (int mm = 0; mm < M_PER_BLOCK; ++mm)
#pragma unroll
            for (int i = 0; i < 8; ++i) {
                s_val[(mm * WAVES_PER_BLOCK + wave) * 16 + rbase + i] = best[mm][i];
                s_idx[(mm * WAVES_PER_BLOCK + wave) * 16 + rbase + i] = bidx[mm][i];
            }
    }
    __syncthreads();

    // Final cross-wave reduce: 32 (m-tile,row) winners, 8 candidates each.
    if (threadIdx.x < 16 * M_PER_BLOCK) {
        const int mm  = threadIdx.x >> 4;
        const int row = threadIdx.x & 15;
        float    bv = -3.0e38f;
        unsigned bi = 0u;
#pragma unroll
        for (int w = 0; w < WAVES_PER_BLOCK; ++w) {
            const float    v  = s_val[(mm * WAVES_PER_BLOCK + w) * 16 + row];
            const unsigned ii = s_idx[(mm * WAVES_PER_BLOCK + w) * 16 + row];
            const bool take = (v > bv) || (v == bv && ii < bi);
            bv = take ? v : bv;
            bi = take ? ii : bi;
        }
        const int qq = (mt0 + mm) * 16 + row;
        idxs[qq] = (long long)bi;
#pragma unroll
        for (int j = 0; j < 8; ++j) vals[qq * 8 + j] = grid[(size_t)bi * 8 + j];
    }
}

// ---------------------------------------------------------------------------
// Kernel 2: _pack_idxs. One thread per packed int64 (2048 total).
// packed flat index f = A*128 + C*32 + B*4 + D (A<16, C<4, B<8, D<4)
// -> 2x2 block a = A*8+B (row pairs 2a,2a+1), b = C*4+D (col pairs 2b,2b+1)
// ---------------------------------------------------------------------------
__global__ __launch_bounds__(256) void e8p_pack(
    const long long* __restrict__ idxs,   // [256, 32]
    long long* __restrict__ packed)       // [2048]
{
    const int f = blockIdx.x * blockDim.x + threadIdx.x;
    if (f >= 2048) return;
    const int A = f >> 7;
    const int C = (f >> 5) & 3;
    const int B = (f >> 2) & 7;
    const int D = f & 3;
    const int a = A * 8 + B;
    const int b = C * 4 + D;

    unsigned w[4];
    w[0] = (unsigned)idxs[(2 * a)     * 32 + 2 * b];
    w[1] = (unsigned)idxs[(2 * a + 1) * 32 + 2 * b];
    w[2] = (unsigned)idxs[(2 * a)     * 32 + 2 * b + 1];
    w[3] = (unsigned)idxs[(2 * a + 1) * 32 + 2 * b + 1];

    const unsigned abs32 = (w[0] >> 8)
                         | ((w[1] >> 8) << 8)
                         | ((w[2] >> 8) << 16)
                         | ((w[3] >> 8) << 24);
    unsigned sign32 = 0u;
#pragma unroll
    for (int e = 0; e < 4; ++e)
#pragma unroll
        for (int j = 0; j < 8; ++j)
            sign32 |= ((w[e] >> j) & 1u) << (4 * j + e);

    packed[f] = (long long)(((unsigned long long)sign32 << 32)
                            | (unsigned long long)abs32);
}

// ---------------------------------------------------------------------------
extern "C" void kernel_launch(void* const* d_in, const int* in_sizes, int n_in,
                              void* d_out, int out_size, void* d_ws, size_t ws_size,
                              hipStream_t stream) {
    (void)in_sizes; (void)n_in; (void)d_ws; (void)ws_size; (void)out_size;

    const float* X     = (const float*)d_in[0];   // (256,32,8) f32
    const float* grid  = (const float*)d_in[1];   // (65536,8)  f32
    const float* gnorm = (const float*)d_in[2];   // (65536,)   f32

    // Outputs concatenated in return order: vals f32 | idxs i64 | packed i64
    float*     vals   = (float*)d_out;
    long long* idxs   = (long long*)(vals + (size_t)NQ * 8);
    long long* packed = idxs + NQ;

    e8p_argmax_wmma<<<NBLOCKS, 256, 0, stream>>>(X, grid, gnorm, vals, idxs);
    e8p_pack<<<2048 / 256, 256, 0, stream>>>(idxs, packed);
}